// TemporalGNN_19971597926555
// MI455X (gfx1250) — compile-verified
//
#include <hip/hip_runtime.h>
#include <math.h>

// Problem constants (match the reference)
#define NN 100000
#define FF 2
#define PP 12
#define HH 32
#define EE 3200000
#define FP (FF * PP)   // 24

typedef float v2f __attribute__((ext_vector_type(2)));
typedef float v8f __attribute__((ext_vector_type(8)));

// Branchless tanh: (e^{2x}-1)/(e^{2x}+1), clamped so e^{2x} never overflows.
__device__ __forceinline__ float tanh_fast(float x) {
    float xc = fminf(fmaxf(x, -15.0f), 15.0f);
    float t = expf(2.0f * xc);
    return (t - 1.0f) / (t + 1.0f);
}
__device__ __forceinline__ float sigmoid_fast(float x) {
    return 1.0f / (1.0f + expf(-x));
}

// ---------------- Kernel 1: zero workspace (agg + deg contiguous) -------------
__global__ void zero_ws_kernel(float* p, int n) {
    int i = blockIdx.x * blockDim.x + threadIdx.x;
    int stride = gridDim.x * blockDim.x;
    for (; i < n; i += stride) p[i] = 0.0f;
}

// ---------------- Kernel 2: degree accumulation ------------------------------
__global__ void degree_kernel(const int* __restrict__ ei, float* __restrict__ deg) {
    int e = blockIdx.x * blockDim.x + threadIdx.x;
    if (e < EE) {
        int c = ei[EE + e];  // col (target)
        atomicAdd(&deg[c], 1.0f);
    }
}

// ---------------- Kernel 3: dis = rsqrt(deg + 1) in place --------------------
__global__ void rsqrt_kernel(float* __restrict__ deg) {
    int n = blockIdx.x * blockDim.x + threadIdx.x;
    if (n < NN) deg[n] = rsqrtf(deg[n] + 1.0f);
}

// ---------------- Kernel 4: edge scatter agg[col] += x[row]*w ----------------
__global__ void scatter_kernel(const float* __restrict__ x,
                               const int* __restrict__ ei,
                               const float* __restrict__ dis,
                               float* __restrict__ agg) {
    int e = blockIdx.x * blockDim.x + threadIdx.x;
    if (e >= EE) return;
    int r = ei[e];
    int c = ei[EE + e];
    float w = dis[r] * dis[c];
    const float4* xr = reinterpret_cast<const float4*>(x) + (size_t)r * 6;  // 24 floats
    float* ac = agg + (size_t)c * FP;
#pragma unroll
    for (int i = 0; i < 6; ++i) {
        float4 v = xr[i];
        atomicAdd(ac + 4 * i + 0, v.x * w);
        atomicAdd(ac + 4 * i + 1, v.y * w);
        atomicAdd(ac + 4 * i + 2, v.z * w);
        atomicAdd(ac + 4 * i + 3, v.w * w);
    }
}

// ---------------- Kernel 5: dense gates with WMMA (1 wave = 16 nodes) --------
__global__ __launch_bounds__(32)
void dense_kernel(const float* __restrict__ x,
                  const float* __restrict__ dis,
                  const float* __restrict__ agg,
                  const float* __restrict__ Wz, const float* __restrict__ bz,
                  const float* __restrict__ Wh, const float* __restrict__ bh,
                  const float* __restrict__ Wlz, const float* __restrict__ blz,
                  const float* __restrict__ Wlh, const float* __restrict__ blh,
                  const float* __restrict__ attn,
                  const float* __restrict__ Wout, const float* __restrict__ bout,
                  float* __restrict__ out) {
    __shared__ float aggL[16][FP];     // self-loop-corrected agg rows
    __shared__ float cz[16][HH];       // conv_z output (A operand staging) / reused for relu(h)
    __shared__ float ch[16][HH];       // conv_h output
    __shared__ float woutL[HH * PP];   // 32x12
    __shared__ float boutL[PP];

    const int lane = threadIdx.x;      // 0..31
    const int half = lane >> 4;        // 0 | 1
    const int l16  = lane & 15;
    const int n0   = blockIdx.x * 16;

    // ---- softmax(attn) constants (12 elems, computed redundantly per lane) ----
    float amax = -INFINITY;
#pragma unroll
    for (int p = 0; p < PP; ++p) amax = fmaxf(amax, attn[p]);
    float asum = 0.0f;
#pragma unroll
    for (int p = 0; p < PP; ++p) asum += expf(attn[p] - amax);
    const float inv_asum = 1.0f / asum;

    // ---- per-lane conv weights (lane == output column) ----
    const float wz0 = Wz[lane], wz1 = Wz[HH + lane], bz_l = bz[lane];
    const float wh0 = Wh[lane], wh1 = Wh[HH + lane], bh_l = bh[lane];
    // gate biases per lane column (tile0: col=l16, tile1: col=l16+16)
    const float blz0 = blz[l16], blz1 = blz[l16 + 16];
    const float blh0 = blh[l16], blh1 = blh[l16 + 16];

    // ---- preload gate weight B-operands in WMMA B layout ----
    // B 4x16 (KxN): VGPR0 -> K=4kk+2*half, VGPR1 -> K=4kk+2*half+1, N = l16 (+16*t)
    v2f bzw[8][2], bhw[8][2];
#pragma unroll
    for (int kk = 0; kk < 8; ++kk) {
        int k0 = 4 * kk + 2 * half;
#pragma unroll
        for (int t = 0; t < 2; ++t) {
            int col = l16 + 16 * t;
            bzw[kk][t] = (v2f){Wlz[k0 * HH + col], Wlz[(k0 + 1) * HH + col]};
            bhw[kk][t] = (v2f){Wlh[k0 * HH + col], Wlh[(k0 + 1) * HH + col]};
        }
    }

    // ---- stage agg rows (with self-loop term) into LDS ----
    for (int i = lane; i < 16 * FP; i += 32) {
        int m = i / FP, k = i % FP;
        int n = n0 + m;
        float d = dis[n];
        aggL[m][k] = agg[(size_t)n * FP + k] + x[(size_t)n * FP + k] * d * d;
    }
    for (int i = lane; i < HH * PP; i += 32) woutL[i] = Wout[i];
    if (lane < PP) boutL[lane] = bout[lane];
    __syncthreads();

    // ---- per-period gate computation, attention-weighted accumulation ----
    v8f hs0 = {}, hs1 = {};   // h_accum tiles (16 nodes x 32 cols)
    for (int p = 0; p < PP; ++p) {
        // conv outputs c_z, c_h: (16 nodes)x(32 cols), lane == col
        for (int m = 0; m < 16; ++m) {
            float a0 = aggL[m][p];
            float a1 = aggL[m][PP + p];
            cz[m][lane] = a0 * wz0 + a1 * wz1 + bz_l;
            ch[m][lane] = a0 * wh0 + a1 * wh1 + bh_l;
        }
        __syncthreads();

        v8f z0 = {}, z1 = {}, h0 = {}, h1 = {};
#pragma unroll
        for (int kk = 0; kk < 8; ++kk) {
            // A 16x4: lane<16 -> row l16, K=4kk..4kk+1 ; lane>=16 -> K=4kk+2..3
            v2f az = *(const v2f*)&cz[l16][4 * kk + 2 * half];
            v2f ah = *(const v2f*)&ch[l16][4 * kk + 2 * half];
            z0 = __builtin_amdgcn_wmma_f32_16x16x4_f32(false, az, false, bzw[kk][0], (short)0, z0, false, false);
            z1 = __builtin_amdgcn_wmma_f32_16x16x4_f32(false, az, false, bzw[kk][1], (short)0, z1, false, false);
            h0 = __builtin_amdgcn_wmma_f32_16x16x4_f32(false, ah, false, bhw[kk][0], (short)0, h0, false, false);
            h1 = __builtin_amdgcn_wmma_f32_16x16x4_f32(false, ah, false, bhw[kk][1], (short)0, h1, false, false);
        }

        float prob = expf(attn[p] - amax) * inv_asum;
#pragma unroll
        for (int v = 0; v < 8; ++v) {
            float zz0 = sigmoid_fast(z0[v] + blz0);
            float zz1 = sigmoid_fast(z1[v] + blz1);
            float th0 = tanh_fast(h0[v] + blh0);
            float th1 = tanh_fast(h1[v] + blh1);
            hs0[v] += prob * (1.0f - zz0) * th0;
            hs1[v] += prob * (1.0f - zz1) * th1;
        }
        __syncthreads();  // before next period overwrites cz/ch
    }

    // ---- relu(h_accum) back to LDS (D layout: lane holds rows M=v+8*half, col=l16(+16)) ----
#pragma unroll
    for (int v = 0; v < 8; ++v) {
        cz[v + 8 * half][l16]      = fmaxf(hs0[v], 0.0f);
        cz[v + 8 * half][l16 + 16] = fmaxf(hs1[v], 0.0f);
    }
    __syncthreads();

    // ---- output matmul (16x32)@(32x12): 192 outputs, 6 per lane ----
#pragma unroll
    for (int j = 0; j < 6; ++j) {
        int idx = lane * 6 + j;
        int m = idx / PP, q = idx % PP;
        float s = boutL[q];
#pragma unroll
        for (int k = 0; k < HH; ++k) s += cz[m][k] * woutL[k * PP + q];
        out[(size_t)(n0 + m) * PP + q] = s;
    }
}

// ---------------------------------------------------------------------------
extern "C" void kernel_launch(void* const* d_in, const int* in_sizes, int n_in,
                              void* d_out, int out_size, void* d_ws, size_t ws_size,
                              hipStream_t stream) {
    (void)in_sizes; (void)n_in; (void)out_size; (void)ws_size;
    const float* x    = (const float*)d_in[0];
    const int*   ei   = (const int*)  d_in[1];
    const float* Wz   = (const float*)d_in[2];
    const float* bz   = (const float*)d_in[3];
    // d_in[4], d_in[5] (Wr, br) unused: r-gate is dead (H0 == 0)
    const float* Wh   = (const float*)d_in[6];
    const float* bh   = (const float*)d_in[7];
    const float* Wlz  = (const float*)d_in[8];
    const float* blz  = (const float*)d_in[9];
    // d_in[10], d_in[11] (Wlr, blr) unused
    const float* Wlh  = (const float*)d_in[12];
    const float* blh  = (const float*)d_in[13];
    const float* attn = (const float*)d_in[14];
    const float* Wout = (const float*)d_in[15];
    const float* bout = (const float*)d_in[16];
    float* out = (float*)d_out;

    // Workspace layout: [agg: NN*24 f32][deg/dis: NN f32]  (= 10 MB)
    float* agg = (float*)d_ws;
    float* deg = agg + (size_t)NN * FP;

    // 1) zero agg + deg (contiguous)
    {
        int n = NN * (FP + 1);
        int blocks = min((n + 255) / 256, 4096);
        zero_ws_kernel<<<blocks, 256, 0, stream>>>(agg, n);
    }
    // 2) degrees
    degree_kernel<<<(EE + 255) / 256, 256, 0, stream>>>(ei, deg);
    // 3) dis = rsqrt(deg + 1)
    rsqrt_kernel<<<(NN + 255) / 256, 256, 0, stream>>>(deg);
    // 4) edge scatter
    scatter_kernel<<<(EE + 255) / 256, 256, 0, stream>>>(x, ei, deg, agg);
    // 5) dense WMMA phase: one wave per 16 nodes (NN divisible by 16)
    dense_kernel<<<NN / 16, 32, 0, stream>>>(x, deg, agg, Wz, bz, Wh, bh,
                                             Wlz, blz, Wlh, blh, attn,
                                             Wout, bout, out);
}